// CombPool2d_7980049236293
// MI455X (gfx1250) — compile-verified
//
#include <hip/hip_runtime.h>
#include <stdint.h>

// CombPool2d: out = w_avg^2 * avgpool2x2(x) + w_max^2 * maxpool2x2(x)
// x: (16,192,224,224) f32 NCHW, stride-2 2x2 window -> out (16,192,112,112)
//
// Memory-bound streaming op (0.5 flop/byte -> HBM roofline ~33us). MI455X strategy:
//  - per-lane global_load_async_to_lds_b128 (ASYNCcnt) for deep prefetch,
//    th:TH_LOAD_NT so the 616MB one-shot input stream doesn't rinse the 192MB L2
//  - staged s_wait_asynccnt (6/4/2/0): consume stage k while later stages in flight
//  - ds_load_b128 consume from LDS (lane-private slots -> no barrier needed)
//  - nontemporal b64 stores for the 154MB one-shot output stream

typedef float v4f __attribute__((ext_vector_type(4)));
typedef float v2f __attribute__((ext_vector_type(2)));

#define TPB   256   // 8 waves (wave32)
#define ITERS 4     // pairs per thread, grid-interleaved

template <bool EXACT>
__global__ __launch_bounds__(TPB) void combpool2x2_kernel(
    const float* __restrict__ x,
    const float* __restrict__ w_avg,
    const float* __restrict__ w_max,
    float* __restrict__ out,
    unsigned int npairs,
    unsigned int stride_pairs)
{
    // 4 stages x 256 threads x 2 row-slots x 16B = 32 KB LDS (~10 blocks/WGP of 320KB)
    __shared__ v4f buf[ITERS][TPB][2];

    const unsigned int tid = threadIdx.x;
    const unsigned int p0  = blockIdx.x * TPB + tid;

    unsigned int chs[ITERS];   // per-stage channel index, reused in phase 2

    // ---- Phase 1: issue ALL async global->LDS loads (8 x b128 per thread) ----
#pragma unroll
    for (int k = 0; k < ITERS; ++k) {
        unsigned int p = p0 + (unsigned int)k * stride_pairs;
        if (EXACT || p < npairs) {
            // p indexes a horizontal pair of outputs: 56 pairs per output row
            unsigned int ow2   = p % 56u;          // pair column
            unsigned int R     = p / 56u;          // global output row
            unsigned int oh    = R % 112u;
            unsigned int plane = R / 112u;         // b*192 + c
            chs[k] = plane % 192u;
            const float* r0 = x + (size_t)(plane * 224u + 2u * oh) * 224u + 4u * ow2;
            const float* r1 = r0 + 224;
            // low 32 bits of the generic pointer to an LDS object == LDS byte address
            unsigned int l0 = (unsigned int)(uintptr_t)&buf[k][tid][0];
            unsigned int l1 = (unsigned int)(uintptr_t)&buf[k][tid][1];
            asm volatile("global_load_async_to_lds_b128 %0, %1, off th:TH_LOAD_NT"
                         :: "v"(l0), "v"(r0) : "memory");
            asm volatile("global_load_async_to_lds_b128 %0, %1, off th:TH_LOAD_NT"
                         :: "v"(l1), "v"(r1) : "memory");
        }
    }

    if (!EXACT) {
        // guarded issue -> per-wave ASYNCcnt is data-dependent; wait for everything
        asm volatile("s_wait_asynccnt 0x0" ::: "memory");
    }

    // ---- Phase 2: consume from LDS, compute, NT-store ----
#pragma unroll
    for (int k = 0; k < ITERS; ++k) {
        if (EXACT) {
            // all waves issued exactly 8 async loads, completing in order:
            // stage k is done once outstanding <= 2*(ITERS-1-k)
            if (k == 0)      asm volatile("s_wait_asynccnt 0x6" ::: "memory");
            else if (k == 1) asm volatile("s_wait_asynccnt 0x4" ::: "memory");
            else if (k == 2) asm volatile("s_wait_asynccnt 0x2" ::: "memory");
            else             asm volatile("s_wait_asynccnt 0x0" ::: "memory");
        }
        unsigned int p = p0 + (unsigned int)k * stride_pairs;
        if (EXACT || p < npairs) {
            v4f a = buf[k][tid][0];   // row 2*oh,   cols 4*ow2 .. 4*ow2+3
            v4f b = buf[k][tid][1];   // row 2*oh+1, cols 4*ow2 .. 4*ow2+3

            float wa = w_avg[chs[k]]; wa *= wa;   // 192-float tables: L2-resident
            float wm = w_max[chs[k]]; wm *= wm;

            float avg0 = (a.x + a.y + b.x + b.y) * 0.25f;
            float mx0  = fmaxf(fmaxf(a.x, a.y), fmaxf(b.x, b.y));
            float avg1 = (a.z + a.w + b.z + b.w) * 0.25f;
            float mx1  = fmaxf(fmaxf(a.z, a.w), fmaxf(b.z, b.w));

            v2f o;
            o.x = wa * avg0 + wm * mx0;
            o.y = wa * avg1 + wm * mx1;
            __builtin_nontemporal_store(o, (v2f*)(out + 2ull * p));
        }
    }
}

extern "C" void kernel_launch(void* const* d_in, const int* in_sizes, int n_in,
                              void* d_out, int out_size, void* d_ws, size_t ws_size,
                              hipStream_t stream) {
    (void)in_sizes; (void)n_in; (void)d_ws; (void)ws_size;
    const float* x  = (const float*)d_in[0];
    const float* wa = (const float*)d_in[1];
    const float* wm = (const float*)d_in[2];
    float* out = (float*)d_out;

    unsigned int npairs = (unsigned int)(out_size / 2);           // 19,267,584
    unsigned int total_threads = (npairs + ITERS - 1) / ITERS;    // 4,816,896
    unsigned int blocks = (total_threads + TPB - 1) / TPB;        // 18,816 (exact, no tail)
    unsigned int stride = blocks * TPB;

    if (npairs % (TPB * ITERS) == 0) {
        combpool2x2_kernel<true><<<blocks, TPB, 0, stream>>>(x, wa, wm, out, npairs, stride);
    } else {
        combpool2x2_kernel<false><<<blocks, TPB, 0, stream>>>(x, wa, wm, out, npairs, stride);
    }
}